// stgnn_83975200572034
// MI455X (gfx1250) — compile-verified
//
#include <hip/hip_runtime.h>
#include <math.h>

// ---- model dims ----
#define Bq     64
#define Nn     127
#define NP     128   // padded adjacency row stride (col 127 zeroed)
#define INDIM  100
#define TIN    5
#define RC     32
#define CC     32
#define SC     64
#define EC     128
#define OUTC   12
#define NODED  40
#define SUBK   30
#define RFt    19

typedef float v8f __attribute__((ext_vector_type(8)));
typedef float v2f __attribute__((ext_vector_type(2)));

__device__ __forceinline__ float sigm(float x) { return 1.0f / (1.0f + expf(-x)); }
__device__ __forceinline__ int imin(int a, int b) { return a < b ? a : b; }
__device__ __forceinline__ int imax(int a, int b) { return a > b ? a : b; }

__device__ __forceinline__ v8f wmma4(v2f a, v2f b, v8f c) {
  // V_WMMA_F32_16X16X4_F32: D = A(16x4) x B(4x16) + C(16x16)
  return __builtin_amdgcn_wmma_f32_16x16x4_f32(false, a, false, b, (short)0, c, false, false);
}

// ---------------- WMMA GEMM kernels (branchless inner loops) ----------------
// Rows/cols outside range are computed from clamped (in-bounds) addresses and
// simply never stored; only the K dimension needs exact semantics.

// conv1x1 as GEMM: out[b,co,p] = act( sum_c W[co,c]*in[b,c,p] + bias[co] [+ out_old] )
__global__ void __launch_bounds__(32)
k_conv1x1(const float* __restrict__ in, const float* __restrict__ W,
          const float* __restrict__ bias, float* __restrict__ out,
          int Cin, int Cout, int P, int act, int accum)
{
  int lane = threadIdx.x;
  int b = blockIdx.z;
  int m0 = blockIdx.y * 16, p0 = blockIdx.x * 16;
  int lm = lane & 15;
  int kh = (lane >> 4) * 2;
  int m = m0 + lm;
  int p = p0 + lm;
  int mc = imin(m, Cout - 1);
  int pc = imin(p, P - 1);
  const float* Wr   = W + (long)mc * Cin + kh;           // 8B aligned: Cin even, kh even
  const float* bptr = in + (long)b * Cin * P + pc + (long)kh * P;
  v8f acc = {};
  for (int k = 0; k < Cin; k += 4) {
    float2 aw = *(const float2*)(Wr + k);
    v2f av, bv;
    av.x = aw.x; av.y = aw.y;
    bv.x = bptr[0];
    bv.y = bptr[P];
    bptr += 4 * (long)P;
    acc = wmma4(av, bv, acc);
  }
  int rb = (lane >> 4) * 8;
  for (int v = 0; v < 8; ++v) {
    int mm = m0 + rb + v;
    if (mm < Cout && p < P) {
      float r = acc[v] + bias[mm];
      float* op = out + (long)b * Cout * P + (long)mm * P + p;
      if (accum) r += *op;
      if (act == 1) r = sigm(r);
      else if (act == 2) r = fmaxf(r, 0.f);
      else if (act == 3) r = tanhf(r);
      *op = r;
    }
  }
}

// temporal conv as im2col GEMM over Kdim=Cin*Kt; input [B,Cin,N,Tin] -> out [B,Cout,N,Tout]
// input time index ti = t + k - tpad; zeros outside [0,Tin) (real padding math -> mask)
__global__ void __launch_bounds__(32)
k_tconv(const float* __restrict__ in, const float* __restrict__ W,
        const float* __restrict__ bias, float* __restrict__ out,
        int Cin, int Kt, int Cout, int Tin, int Tout, int tpad, int act, int accum)
{
  int lane = threadIdx.x;
  int b = blockIdx.z;
  int m0 = blockIdx.y * 16, q0 = blockIdx.x * 16;
  int lm = lane & 15, kh = (lane >> 4) * 2;
  int m = m0 + lm;
  int q = q0 + lm;
  int P = Nn * Tout;
  int qc = imin(q, P - 1);
  int n = qc / Tout, tt = qc % Tout;
  int Kdim = Cin * Kt;
  int mc = imin(m, Cout - 1);
  const float* Wr  = W + (long)mc * Kdim + kh;           // 8B aligned: Kdim even, kh even
  const float* inb = in + (long)b * Cin * Nn * Tin + n * Tin;
  v8f acc = {};
  for (int r = 0; r < Kdim; r += 4) {
    int ra = r + kh;
    float2 aw = *(const float2*)(Wr + r);
    v2f av, bv;
    int c0 = ra / Kt, k0 = ra - c0 * Kt;
    int ti0 = tt + k0 - tpad;
    float msk0 = (ti0 >= 0 && ti0 < Tin) ? 1.f : 0.f;
    int tc0 = imin(imax(ti0, 0), Tin - 1);
    int ra1 = ra + 1;
    int c1 = ra1 / Kt, k1 = ra1 - c1 * Kt;
    int ti1 = tt + k1 - tpad;
    float msk1 = (ti1 >= 0 && ti1 < Tin) ? 1.f : 0.f;
    int tc1 = imin(imax(ti1, 0), Tin - 1);
    av.x = aw.x; av.y = aw.y;
    bv.x = inb[c0 * Nn * Tin + tc0] * msk0;
    bv.y = inb[c1 * Nn * Tin + tc1] * msk1;
    acc = wmma4(av, bv, acc);
  }
  int rb = (lane >> 4) * 8;
  for (int v = 0; v < 8; ++v) {
    int mm = m0 + rb + v;
    if (mm < Cout && q < P) {
      float r = acc[v] + bias[mm];
      float* op = out + (long)b * Cout * P + (long)mm * P + q;
      if (accum) r += *op;
      if (act == 1) r = sigm(r);
      else if (act == 2) r = fmaxf(r, 0.f);
      else if (act == 3) r = tanhf(r);
      *op = r;
    }
  }
}

// graph propagation: out[b,c,v,t] = alpha*x[b,c,v,t] + beta * sum_w a[v,w]*in[b,c,w,t]
// amat has row stride NP=128 with column 127 zeroed -> K padded to 128 with no masks.
__global__ void __launch_bounds__(32)
k_nconv(const float* __restrict__ amat,
        const float* __restrict__ in, long inB,
        const float* __restrict__ xsrc, long xB,
        float* __restrict__ out, long outB,
        int C, int T, float alpha, float beta)
{
  int lane = threadIdx.x;
  int b = blockIdx.z;
  int v0 = blockIdx.y * 16, q0 = blockIdx.x * 16;
  int lm = lane & 15, kh = (lane >> 4) * 2;
  int vr = v0 + lm;
  int q = q0 + lm;
  int CT = C * T;
  int qc = imin(q, CT - 1);
  int c = qc / T, tt = qc - c * T;
  int vrc = imin(vr, Nn - 1);
  const float* arow = amat + vrc * NP + kh;              // 8B aligned (NP even, kh even)
  const float* hcol = in + (long)b * inB + (long)c * Nn * T + tt;
  v8f acc = {};
  for (int k = 0; k < NP; k += 4) {
    int ka = k + kh;
    float2 aw = *(const float2*)(arow + k);
    v2f av, bv;
    av.x = aw.x; av.y = aw.y;                            // col 127 is zero -> kills B row 127
    bv.x = hcol[imin(ka, Nn - 1) * T];
    bv.y = hcol[imin(ka + 1, Nn - 1) * T];
    acc = wmma4(av, bv, acc);
  }
  int rb = (lane >> 4) * 8;
  for (int v = 0; v < 8; ++v) {
    int vv = v0 + rb + v;
    if (vv < Nn && q < CT) {
      long rel = (long)c * Nn * T + vv * T + tt;
      out[(long)b * outB + rel] = alpha * xsrc[(long)b * xB + rel] + beta * acc[v];
    }
  }
}

// gram[b,n,m] = sum_o z[b,o,n,t] * z[b,o,m,t]  (z layout [B,RC,N,T]); K=32 exact
__global__ void __launch_bounds__(32)
k_gram(const float* __restrict__ z, float* __restrict__ gram, int T, int t)
{
  int lane = threadIdx.x;
  int b = blockIdx.z;
  int n0 = blockIdx.y * 16, m0 = blockIdx.x * 16;
  int lm = lane & 15, kh = (lane >> 4) * 2;
  int nr = imin(n0 + lm, Nn - 1);
  int mc = imin(m0 + lm, Nn - 1);
  const float* zb = z + (long)b * RC * Nn * T + t;
  const float* za = zb + nr * T + (long)kh * Nn * T;
  const float* zB = zb + mc * T + (long)kh * Nn * T;
  long stride = (long)Nn * T;
  v8f acc = {};
  for (int k = 0; k < RC; k += 4) {
    v2f av, bv;
    av.x = za[0]; av.y = za[stride];
    bv.x = zB[0]; bv.y = zB[stride];
    za += 4 * stride; zB += 4 * stride;
    acc = wmma4(av, bv, acc);
  }
  int rb = (lane >> 4) * 8;
  int mcol = m0 + lm;
  for (int v = 0; v < 8; ++v) {
    int nn = n0 + rb + v;
    if (nn < Nn && mcol < Nn)
      gram[(long)b * Nn * Nn + nn * Nn + mcol] = acc[v];
  }
}

// ---------------- scalar / elementwise kernels ----------------

__global__ void k_nv(const float* __restrict__ emb, const float* __restrict__ W,
                     const float* __restrict__ bias, float* __restrict__ out)
{
  int i = blockIdx.x * blockDim.x + threadIdx.x;
  if (i >= Nn * NODED) return;
  int r = i / NODED, c = i % NODED;
  float s = bias[c];
  for (int k = 0; k < NODED; ++k) s += emb[r * NODED + k] * W[c * NODED + k];
  out[i] = tanhf(3.0f * s);
}

__global__ void k_adjraw(const float* __restrict__ nv1, const float* __restrict__ nv2,
                         float* __restrict__ adj)
{
  int i = blockIdx.x * blockDim.x + threadIdx.x;
  if (i >= Nn * Nn) return;
  int r = i / Nn, c = i % Nn;
  float s = 0.f;
  for (int k = 0; k < NODED; ++k)
    s += nv1[r * NODED + k] * nv2[c * NODED + k] - nv2[r * NODED + k] * nv1[c * NODED + k];
  adj[i] = fmaxf(tanhf(3.0f * s), 0.f);
}

__global__ void k_topk(const float* __restrict__ adj, float* __restrict__ adp)
{
  int i = blockIdx.x * blockDim.x + threadIdx.x;
  if (i >= Nn) return;
  unsigned pick[4] = {0u, 0u, 0u, 0u};
  for (int j = 0; j < Nn; ++j) adp[i * Nn + j] = 0.f;
  for (int s = 0; s < SUBK; ++s) {
    float best = -1e30f; int bj = 0;
    for (int j = 0; j < Nn; ++j) {
      if (pick[j >> 5] & (1u << (j & 31))) continue;
      float v = adj[i * Nn + j];
      if (v > best) { best = v; bj = j; }
    }
    pick[bj >> 5] |= 1u << (bj & 31);
    adp[i * Nn + bj] = best;
  }
}

// a1[v,w]=(adp[v,w]+d)/(1+rowsum_v) ; a2[v,w]=(adp[w,v]+d)/(1+colsum_v); stride NP, col 127 zero
__global__ void k_norm_adp(const float* __restrict__ adp, float* __restrict__ a1,
                           float* __restrict__ a2)
{
  __shared__ float r1[128], r2[128];
  int v = blockIdx.x, tid = threadIdx.x;
  float s1 = 0.f, s2 = 0.f;
  if (tid < Nn) { s1 = adp[v * Nn + tid]; s2 = adp[tid * Nn + v]; }
  r1[tid] = s1; r2[tid] = s2; __syncthreads();
  for (int st = 64; st > 0; st >>= 1) {
    if (tid < st) { r1[tid] += r1[tid + st]; r2[tid] += r2[tid + st]; }
    __syncthreads();
  }
  float S1 = r1[0] + 1.f, S2 = r2[0] + 1.f;
  if (tid < Nn) {
    float d = (tid == v) ? 1.f : 0.f;
    a1[v * NP + tid] = (adp[v * Nn + tid] + d) / S1;
    a2[v * NP + tid] = (adp[tid * Nn + v] + d) / S2;
  } else {  // tid == 127: zero the pad column
    a1[v * NP + 127] = 0.f;
    a2[v * NP + 127] = 0.f;
  }
}

// column-normalized A_t + I : block per column w; s = 1 + rowsum(row w of A_t); stride NP out
__global__ void k_norm_at(const float* __restrict__ adjt, int t, float* __restrict__ a)
{
  __shared__ float red[128];
  int w = blockIdx.x, tid = threadIdx.x;
  const float* At = adjt + (long)t * Nn * Nn;
  float s = (tid < Nn) ? At[w * Nn + tid] : 0.f;
  red[tid] = s; __syncthreads();
  for (int st = 64; st > 0; st >>= 1) { if (tid < st) red[tid] += red[tid + st]; __syncthreads(); }
  float sum = red[0] + 1.0f;
  for (int v = tid; v < Nn; v += 128) {
    a[v * NP + w] = (At[v * Nn + w] + (v == w ? 1.f : 0.f)) / sum;
    if (w == 0) a[v * NP + 127] = 0.f;
  }
}

// attention row normalize + mean over batch: adjt[t,n,m] = mean_b e(b,n,m)/(rowsum+1e-6)
__global__ void k_attrow(const float* __restrict__ gram, float* __restrict__ adjt, int t)
{
  __shared__ float red[128];
  int n = blockIdx.x, m = threadIdx.x;
  float accm = 0.f;
  for (int b = 0; b < Bq; ++b) {
    const float* gb = gram + (long)b * Nn * Nn;
    float e = 0.f;
    if (m < Nn) {
      float g   = gb[n * Nn + m];
      float rnn = sqrtf(fmaxf(gb[n * Nn + n], 0.f));
      float rnm = sqrtf(fmaxf(gb[m * Nn + m], 0.f));
      float nn_ = rnn + 1e-6f, nm_ = rnm + 1e-6f;
      float cosv = g / (nn_ * nm_);
      float sq   = rnn * rnn + rnm * rnm - 2.f * g;
      float dist = sqrtf(fmaxf(sq, 1e-12f));
      float fd   = dist / (nn_ + nm_ + 1e-6f);
      e = expf(cosv - fd);
    }
    red[m] = e; __syncthreads();
    for (int s = 64; s > 0; s >>= 1) { if (m < s) red[m] += red[m + s]; __syncthreads(); }
    float ssum = red[0]; __syncthreads();
    if (m < Nn) accm += e / (ssum + 1e-6f);
  }
  if (m < Nn) adjt[(long)t * Nn * Nn + n * Nn + m] = accm / (float)Bq;
}

__global__ void k_fg(float* __restrict__ f, const float* __restrict__ g, int total)
{
  int i = blockIdx.x * blockDim.x + threadIdx.x;
  if (i >= total) return;
  f[i] = tanhf(f[i]) * sigm(g[i]);
}

__global__ void k_resadd(const float* __restrict__ hg, const float* __restrict__ hold,
                         float* __restrict__ hnew, int T1, int T2, int total)
{
  int i = blockIdx.x * blockDim.x + threadIdx.x;
  if (i >= total) return;
  int t = i % T2;
  long rest = i / T2;  // (b*CC + c)*N + n
  hnew[i] = hg[i] + hold[rest * T1 + t + (T1 - T2)];
}

__global__ void k_gate(const float* __restrict__ wx, const float* __restrict__ wh,
                       float* __restrict__ hc, long hcB, float* __restrict__ hsum,
                       int perb, int total)
{
  int i = blockIdx.x * blockDim.x + threadIdx.x;
  if (i >= total) return;
  int b = i / perb, r = i % perb;
  float x = wx[i], h = wh[i];
  float g = sigm(x + h);
  float hn = g * x + (1.f - g) * h;
  hc[(long)b * hcB + r] = hn;
  hsum[i] += hn;
}

__global__ void k_layernorm(const float* __restrict__ in, const float* __restrict__ w,
                            const float* __restrict__ bias, float* __restrict__ out, int M)
{
  __shared__ float rs[256], rq[256];
  int b = blockIdx.x, tid = threadIdx.x;
  const float* xb = in + (long)b * M;
  float s = 0.f, q = 0.f;
  for (int i = tid; i < M; i += 256) { float v = xb[i]; s += v; q += v * v; }
  rs[tid] = s; rq[tid] = q; __syncthreads();
  for (int st = 128; st > 0; st >>= 1) {
    if (tid < st) { rs[tid] += rs[tid + st]; rq[tid] += rq[tid + st]; }
    __syncthreads();
  }
  float mu  = rs[0] / (float)M;
  float var = rq[0] / (float)M - mu * mu;
  float rstd = rsqrtf(var + 1e-5f);
  for (int i = tid; i < M; i += 256)
    out[(long)b * M + i] = (xb[i] - mu) * rstd * w[i] + bias[i];
}

__global__ void k_copyb(const float* __restrict__ src, long srcB, float* __restrict__ dst,
                        long dstB, int perb, int total)
{
  int i = blockIdx.x * blockDim.x + threadIdx.x;
  if (i >= total) return;
  int b = i / perb, r = i % perb;
  dst[(long)b * dstB + r] = src[(long)b * srcB + r];
}

__global__ void k_copy(const float* __restrict__ s, float* __restrict__ d, int n)
{
  int i = blockIdx.x * blockDim.x + threadIdx.x;
  if (i < n) d[i] = s[i];
}

__global__ void k_fill(float* __restrict__ p, float v, int n)
{
  int i = blockIdx.x * blockDim.x + threadIdx.x;
  if (i < n) p[i] = v;
}

__global__ void k_relu(float* __restrict__ p, int n)
{
  int i = blockIdx.x * blockDim.x + threadIdx.x;
  if (i < n) p[i] = fmaxf(p[i], 0.f);
}

// ---------------- host orchestration ----------------

static inline int cdiv(int a, int b) { return (a + b - 1) / b; }

extern "C" void kernel_launch(void* const* d_in, const int* in_sizes, int n_in,
                              void* d_out, int out_size, void* d_ws, size_t ws_size,
                              hipStream_t stream)
{
  (void)in_sizes; (void)n_in; (void)out_size; (void)ws_size;
  int a = 0;
  const float* x        = (const float*)d_in[a++];
  const float* start_W  = (const float*)d_in[a++];
  const float* start_b  = (const float*)d_in[a++];
  const float* skip0_W  = (const float*)d_in[a++];
  const float* skip0_b  = (const float*)d_in[a++];
  const float* filt_W   = (const float*)d_in[a++];
  const float* filt_b   = (const float*)d_in[a++];
  const float* gatec_W  = (const float*)d_in[a++];
  const float* gatec_b  = (const float*)d_in[a++];
  const float* skW0     = (const float*)d_in[a++];
  const float* skb0     = (const float*)d_in[a++];
  const float* skW1     = (const float*)d_in[a++];
  const float* skb1     = (const float*)d_in[a++];
  const float* skW2     = (const float*)d_in[a++];
  const float* skb2     = (const float*)d_in[a++];
  const float* g1_W     = (const float*)d_in[a++];
  const float* g1_b     = (const float*)d_in[a++];
  const float* g2_W     = (const float*)d_in[a++];
  const float* g2_b     = (const float*)d_in[a++];
  const float* g3_W     = (const float*)d_in[a++];
  const float* g3_b     = (const float*)d_in[a++];
  const float* gate_Wx  = (const float*)d_in[a++];
  const float* gate_bx  = (const float*)d_in[a++];
  const float* gate_Wh  = (const float*)d_in[a++];
  const float* gate_bh  = (const float*)d_in[a++];
  const float* norm_w0  = (const float*)d_in[a++];
  const float* norm_b0  = (const float*)d_in[a++];
  const float* norm_w1  = (const float*)d_in[a++];
  const float* norm_b1  = (const float*)d_in[a++];
  const float* norm_w2  = (const float*)d_in[a++];
  const float* norm_b2  = (const float*)d_in[a++];
  const float* gat_W    = (const float*)d_in[a++];
  const float* gat_b    = (const float*)d_in[a++];
  const float* emb1     = (const float*)d_in[a++];
  const float* emb2     = (const float*)d_in[a++];
  const float* lin1_W   = (const float*)d_in[a++];
  const float* lin1_b   = (const float*)d_in[a++];
  const float* lin2_W   = (const float*)d_in[a++];
  const float* lin2_b   = (const float*)d_in[a++];
  const float* skipE_W  = (const float*)d_in[a++];
  const float* skipE_b  = (const float*)d_in[a++];
  const float* end1_W   = (const float*)d_in[a++];
  const float* end1_b   = (const float*)d_in[a++];
  const float* end2_W   = (const float*)d_in[a++];
  const float* end2_b   = (const float*)d_in[a++];

  float* ws = (float*)d_ws;
  size_t off = 0;
  auto wsalloc = [&](size_t n) { float* p = ws + off; off += n; return p; };
  const int TM1 = 19, TM2 = 13;
  float* hA    = wsalloc((size_t)Bq * RC * Nn * TM1);
  float* hB    = wsalloc((size_t)Bq * RC * Nn * TM1);
  float* fb    = wsalloc((size_t)Bq * CC * Nn * TM2);
  float* gb    = wsalloc((size_t)Bq * CC * Nn * TM2);
  float* cat   = wsalloc((size_t)Bq * 3 * CC * Nn * TM2);
  float* xs    = wsalloc((size_t)Bq * RC * Nn * TM2);
  float* wx    = wsalloc((size_t)Bq * RC * Nn * TM2);
  float* xd    = wsalloc((size_t)Bq * RC * Nn * TM2);
  float* wh    = wsalloc((size_t)Bq * RC * Nn * TM2);
  float* hsum  = wsalloc((size_t)Bq * RC * Nn * TM2);
  float* zb    = wsalloc((size_t)Bq * RC * Nn * TM2);
  float* skip  = wsalloc((size_t)Bq * SC * Nn);
  float* gram  = wsalloc((size_t)Bq * Nn * Nn);
  float* adjt  = wsalloc((size_t)TM2 * Nn * Nn);
  float* a1    = wsalloc((size_t)Nn * NP);
  float* a2    = wsalloc((size_t)Nn * NP);
  float* at_   = wsalloc((size_t)Nn * NP);
  float* nv1   = wsalloc((size_t)Nn * NODED);
  float* nv2   = wsalloc((size_t)Nn * NODED);
  float* adjr  = wsalloc((size_t)Nn * Nn);
  float* adp   = wsalloc((size_t)Nn * Nn);
  float* ebuf  = wsalloc((size_t)Bq * EC * Nn);

  dim3 blk32(32);

  // ---- graph construction ----
  k_nv<<<cdiv(Nn * NODED, 128), 128, 0, stream>>>(emb1, lin1_W, lin1_b, nv1);
  k_nv<<<cdiv(Nn * NODED, 128), 128, 0, stream>>>(emb2, lin2_W, lin2_b, nv2);
  k_adjraw<<<cdiv(Nn * Nn, 128), 128, 0, stream>>>(nv1, nv2, adjr);
  k_topk<<<cdiv(Nn, 64), 64, 0, stream>>>(adjr, adp);
  k_copy<<<cdiv(Nn * Nn, 256), 256, 0, stream>>>(adp, (float*)d_out + (size_t)Bq * OUTC * Nn, Nn * Nn);
  k_norm_adp<<<Nn, 128, 0, stream>>>(adp, a1, a2);

  // ---- start conv (pad folded via tpad) and skip0 ----
  {
    dim3 g(cdiv(Nn * RFt, 16), cdiv(RC, 16), Bq);
    k_tconv<<<g, blk32, 0, stream>>>(x, start_W, start_b, hA, INDIM, 1, RC, TIN, RFt, RFt - TIN, 0, 0);
    dim3 g2(cdiv(Nn, 16), cdiv(SC, 16), Bq);
    k_tconv<<<g2, blk32, 0, stream>>>(x, skip0_W, skip0_b, skip, INDIM, RFt, SC, TIN, 1, RFt - TIN, 0, 0);
  }

  const int Tins[3]  = {19, 13, 7};
  const int Touts[3] = {13, 7, 1};
  const float* skWs[3] = {skW0, skW1, skW2};
  const float* skbs[3] = {skb0, skb1, skb2};
  const float* nws[3]  = {norm_w0, norm_w1, norm_w2};
  const float* nbs[3]  = {norm_b0, norm_b1, norm_b2};

  float* hcur = hA;
  float* hnxt = hB;

  for (int L = 0; L < 3; ++L) {
    int T1 = Tins[L], T2 = Touts[L];
    int NT = Nn * T2;
    int total2 = Bq * CC * NT;
    long catB = (long)3 * CC * NT;

    // TCN: filter & gate
    dim3 gt(cdiv(NT, 16), cdiv(CC, 16), Bq);
    k_tconv<<<gt, blk32, 0, stream>>>(hcur, filt_W + (size_t)L * CC * RC * 7, filt_b + L * CC,
                                      fb, RC, 7, CC, T1, T2, 0, 0, 0);
    k_tconv<<<gt, blk32, 0, stream>>>(hcur, gatec_W + (size_t)L * CC * RC * 7, gatec_b + L * CC,
                                      gb, RC, 7, CC, T1, T2, 0, 0, 0);
    k_fg<<<cdiv(total2, 256), 256, 0, stream>>>(fb, gb, total2);

    // skip accumulation (full-time conv -> T=1)
    dim3 gs(cdiv(Nn, 16), cdiv(SC, 16), Bq);
    k_tconv<<<gs, blk32, 0, stream>>>(fb, skWs[L], skbs[L], skip, CC, T2, SC, T2, 1, 0, 0, 1);

    // residual
    k_resadd<<<cdiv(total2, 256), 256, 0, stream>>>(fb, hcur, hnxt, T1, T2, total2);
    { float* t = hcur; hcur = hnxt; hnxt = t; }  // hcur = post-residual h, len T2

    // ---- x_s = mixprop(h,a1) + mixprop(h,a2) ----
    k_copyb<<<cdiv(total2, 256), 256, 0, stream>>>(hcur, (long)CC * NT, cat, catB, CC * NT, total2);
    dim3 gn(cdiv(CC * T2, 16), cdiv(Nn, 16), Bq);
    dim3 gc1(cdiv(NT, 16), cdiv(RC, 16), Bq);
    k_nconv<<<gn, blk32, 0, stream>>>(a1, cat, catB, cat, catB, cat + CC * NT, catB, CC, T2, 0.05f, 0.95f);
    k_nconv<<<gn, blk32, 0, stream>>>(a1, cat + CC * NT, catB, cat, catB, cat + 2 * CC * NT, catB, CC, T2, 0.05f, 0.95f);
    k_conv1x1<<<gc1, blk32, 0, stream>>>(cat, g1_W + (size_t)L * RC * 3 * CC, g1_b + L * RC,
                                         xs, 3 * CC, RC, NT, 0, 0);
    k_nconv<<<gn, blk32, 0, stream>>>(a2, cat, catB, cat, catB, cat + CC * NT, catB, CC, T2, 0.05f, 0.95f);
    k_nconv<<<gn, blk32, 0, stream>>>(a2, cat + CC * NT, catB, cat, catB, cat + 2 * CC * NT, catB, CC, T2, 0.05f, 0.95f);
    k_conv1x1<<<gc1, blk32, 0, stream>>>(cat, g2_W + (size_t)L * RC * 3 * CC, g2_b + L * RC,
                                         xs, 3 * CC, RC, NT, 0, 1);

    // wx = conv1x1(x_s) (loop-invariant over scan)
    k_conv1x1<<<gc1, blk32, 0, stream>>>(xs, gate_Wx + (size_t)L * RC * RC, gate_bx + L * RC,
                                         wx, RC, RC, NT, 0, 0);

    // ---- GAT attention adjacency per time step ----
    k_conv1x1<<<gc1, blk32, 0, stream>>>(hcur, gat_W, gat_b, zb, RC, RC, NT, 3, 0);
    for (int t = 0; t < T2; ++t) {
      dim3 gg(cdiv(Nn, 16), cdiv(Nn, 16), Bq);
      k_gram<<<gg, blk32, 0, stream>>>(zb, gram, T2, t);
      k_attrow<<<Nn, 128, 0, stream>>>(gram, adjt, t);
    }

    // ---- sequential scan over time; carry hc lives in cat[:, 0:32] ----
    k_fill<<<cdiv(total2, 256), 256, 0, stream>>>(hsum, 0.f, total2);
    for (int t = 0; t < T2; ++t) {
      k_norm_at<<<Nn, 128, 0, stream>>>(adjt, t, at_);
      k_nconv<<<gn, blk32, 0, stream>>>(at_, cat, catB, cat, catB, cat + CC * NT, catB, CC, T2, 0.05f, 1.0f);
      k_nconv<<<gn, blk32, 0, stream>>>(at_, cat + CC * NT, catB, cat, catB, cat + 2 * CC * NT, catB, CC, T2, 0.05f, 1.0f);
      k_conv1x1<<<gc1, blk32, 0, stream>>>(cat, g3_W + (size_t)L * RC * 3 * CC, g3_b + L * RC,
                                           xd, 3 * CC, RC, NT, 1, 0);  // sigmoid
      k_conv1x1<<<gc1, blk32, 0, stream>>>(xd, gate_Wh + (size_t)L * RC * RC, gate_bh + L * RC,
                                           wh, RC, RC, NT, 0, 0);
      k_gate<<<cdiv(total2, 256), 256, 0, stream>>>(wx, wh, cat, catB, hsum, CC * NT, total2);
    }

    // layernorm(sum of scan outputs) -> next h
    k_layernorm<<<Bq, 256, 0, stream>>>(hsum, nws[L], nbs[L], hnxt, CC * NT);
    { float* t = hcur; hcur = hnxt; hnxt = t; }
  }

  // ---- output head ----
  dim3 ge1(cdiv(Nn, 16), cdiv(SC, 16), Bq);
  k_conv1x1<<<ge1, blk32, 0, stream>>>(hcur, skipE_W, skipE_b, skip, RC, SC, Nn, 0, 1);
  k_relu<<<cdiv(Bq * SC * Nn, 256), 256, 0, stream>>>(skip, Bq * SC * Nn);
  dim3 ge2(cdiv(Nn, 16), cdiv(EC, 16), Bq);
  k_conv1x1<<<ge2, blk32, 0, stream>>>(skip, end1_W, end1_b, ebuf, SC, EC, Nn, 2, 0);
  dim3 ge3(cdiv(Nn, 16), cdiv(OUTC, 16), Bq);
  k_conv1x1<<<ge3, blk32, 0, stream>>>(ebuf, end2_W, end2_b, (float*)d_out, EC, OUTC, Nn, 0, 0);
}